// SpilloverCompensationLayer_75814762709166
// MI455X (gfx1250) — compile-verified
//
#include <hip/hip_runtime.h>
#include <hip/hip_bf16.h>

typedef float v2f __attribute__((ext_vector_type(2)));
typedef float v8f __attribute__((ext_vector_type(8)));
typedef int   v2i __attribute__((ext_vector_type(2)));

#define D_FEAT 128
#define COL_PAD 16

// ---------------- Phase 0: zero degree + cursor + col_s tail padding ----------------
__global__ void init_counts_kernel(int* __restrict__ deg, int* __restrict__ cur,
                                   int* __restrict__ col_s, int n, int e) {
    int i = blockIdx.x * blockDim.x + threadIdx.x;
    if (i < n) { deg[i] = 0; cur[i] = 0; }
    if (i < COL_PAD) col_s[e + i] = 0;   // safe zero pad for unconditional window loads
}

// ---------------- Phase 1: per-destination edge counts ----------------
__global__ void count_kernel(const long long* __restrict__ row, int* __restrict__ deg, int e) {
    int i = blockIdx.x * blockDim.x + threadIdx.x;
    if (i < e) atomicAdd(&deg[(int)row[i]], 1);
}

// ---------------- Phase 2: exclusive scan (single block, LDS) ----------------
__global__ __launch_bounds__(1024)
void scan_kernel(const int* __restrict__ deg, int* __restrict__ off, int n) {
    __shared__ int sdata[1024];
    __shared__ int carry;
    int tid = threadIdx.x;
    if (tid == 0) { carry = 0; off[0] = 0; }
    __syncthreads();
    for (int base = 0; base < n; base += 1024) {
        int i = base + tid;
        int v = (i < n) ? deg[i] : 0;
        sdata[tid] = v;
        __syncthreads();
        for (int s = 1; s < 1024; s <<= 1) {
            int t = (tid >= s) ? sdata[tid - s] : 0;
            __syncthreads();
            sdata[tid] += t;
            __syncthreads();
        }
        if (i < n) off[i + 1] = carry + sdata[tid];
        __syncthreads();
        if (tid == 1023) carry += sdata[1023];
        __syncthreads();
    }
}

// ---------------- Phase 3: bucket-fill columns (counting sort by dest) ----------------
__global__ void fill_kernel(const long long* __restrict__ row, const long long* __restrict__ col,
                            const int* __restrict__ off, int* __restrict__ cur,
                            int* __restrict__ col_s, int e) {
    int i = blockIdx.x * blockDim.x + threadIdx.x;
    if (i < e) {
        int r = (int)row[i];
        int p = atomicAdd(&cur[r], 1);
        col_s[off[r] + p] = (int)col[i];
    }
}

// ---------------- Phase 4: WMMA segment-sum + fused finalize ----------------
// One wave per (16-node tile, 32-feature slab = two 16-feature WMMA tiles).
// D = A(16 nodes x 4 edges, 0/1 CSR indicator) * B(4 edges x 16 feats), accumulated.
__global__ __launch_bounds__(128)
void gather_wmma_kernel(const float* __restrict__ x, const int* __restrict__ off,
                        const int* __restrict__ col_s, const float* __restrict__ rate,
                        float* __restrict__ out, int n) {
    const int lane  = threadIdx.x;             // 0..31 (one wave32)
    const int nb    = (int)blockIdx.x * 16;    // node tile base
    const int m     = lane & 15;               // node row (A) / feature col (B,C)
    const int khalf = (lane >> 4) << 1;        // K offset: 0 (lanes 0-15) / 2 (lanes 16-31)
    const unsigned f0 = (unsigned)threadIdx.y * 32u + (unsigned)m;  // feature tile 0 col
    // feature tile 1 col = f0 + 16 (folded into load offset:64)

    // Per-lane CSR range of node nb+m (branch-free clamp)
    int node_m = nb + m;
    int nm = (node_m < n) ? node_m : 0;
    int offS = off[nm];
    unsigned lenm = (node_m < n) ? (unsigned)(off[nm + 1] - offS) : 0u;

    const int tileS = off[nb];
    const int tileE = off[(nb + 16 < n) ? (nb + 16) : n];
    const int ebS   = tileS & ~3;   // 4-aligned window start; pre-tile edges get A=0

    v8f acc0 = {};
    v8f acc1 = {};

    // ---- 2-deep software pipeline; 32-bit element offsets (x < 2^31 elements) ----
    v2i cA = *(const v2i*)(col_s + ebS + khalf);
    v2i cB = *(const v2i*)(col_s + ebS + 4 + khalf);
    unsigned oA0 = ((unsigned)cA.x << 7) + f0;
    unsigned oA1 = ((unsigned)cA.y << 7) + f0;
    float b0a = x[oA0];
    float b1a = x[oA1];
    float b0b = x[oA0 + 16u];
    float b1b = x[oA1 + 16u];

    for (int eb = ebS; eb < tileE; eb += 4) {
        // issue cols for window +2
        v2i cC = *(const v2i*)(col_s + eb + 8 + khalf);
        // issue features for window +1 (cols loaded last iteration)
        unsigned o0 = ((unsigned)cB.x << 7) + f0;
        unsigned o1 = ((unsigned)cB.y << 7) + f0;
        float nb0a = x[o0];
        float nb1a = x[o1];
        float nb0b = x[o0 + 16u];
        float nb1b = x[o1 + 16u];

        // 0/1 indicator for current window (unsigned-range trick)
        int e0 = eb + khalf;
        float a0 = ((unsigned)(e0     - offS) < lenm) ? 1.0f : 0.0f;
        float a1 = ((unsigned)(e0 + 1 - offS) < lenm) ? 1.0f : 0.0f;
        v2f A  = {a0, a1};
        v2f B0 = {b0a, b1a};
        v2f B1 = {b0b, b1b};
        acc0 = __builtin_amdgcn_wmma_f32_16x16x4_f32(false, A, false, B0,
                                                     (short)0, acc0, false, false);
        acc1 = __builtin_amdgcn_wmma_f32_16x16x4_f32(false, A, false, B1,
                                                     (short)0, acc1, false, false);
        b0a = nb0a; b1a = nb1a; b0b = nb0b; b1b = nb1b;
        cB = cC;
    }

    // Fused finalize: out = relu(x - r * (sum / max(deg,1)))
    float r = rate[0];
    r = fminf(fmaxf(r, 0.0f), 0.2f);
    const int mrow0 = (lane >> 4) * 8;  // C layout: lanes 0-15 -> M=rr, lanes 16-31 -> M=rr+8

    if (nb + 16 <= n) {
        // Uniform fast path: no per-row guards, no exec churn
        #pragma unroll
        for (int rr = 0; rr < 8; ++rr) {
            unsigned node = (unsigned)(nb + rr + mrow0);
            float degv = (float)(off[node + 1] - off[node]);
            degv = fmaxf(degv, 1.0f);
            float scale = r * __builtin_amdgcn_rcpf(degv);
            unsigned o = (node << 7) + f0;
            float xv0 = x[o];
            float xv1 = x[o + 16u];
            out[o]       = fmaxf(xv0 - scale * acc0[rr], 0.0f);
            out[o + 16u] = fmaxf(xv1 - scale * acc1[rr], 0.0f);
        }
    } else {
        for (int rr = 0; rr < 8; ++rr) {
            int node = nb + rr + mrow0;
            if (node >= n) continue;
            float degv = (float)(off[node + 1] - off[node]);
            degv = fmaxf(degv, 1.0f);
            float scale = r * __builtin_amdgcn_rcpf(degv);
            unsigned o = ((unsigned)node << 7) + f0;
            float xv0 = x[o];
            float xv1 = x[o + 16u];
            out[o]       = fmaxf(xv0 - scale * acc0[rr], 0.0f);
            out[o + 16u] = fmaxf(xv1 - scale * acc1[rr], 0.0f);
        }
    }
}

extern "C" void kernel_launch(void* const* d_in, const int* in_sizes, int n_in,
                              void* d_out, int out_size, void* d_ws, size_t ws_size,
                              hipStream_t stream) {
    const float*     x    = (const float*)d_in[0];
    const long long* ei   = (const long long*)d_in[1];
    const float*     rate = (const float*)d_in[2];
    float*           out  = (float*)d_out;

    const int N = in_sizes[0] / D_FEAT;   // 40000
    const int E = in_sizes[1] / 2;        // 640000
    const long long* row = ei;
    const long long* col = ei + E;

    // Workspace carve-up (ints), 256B-aligned chunks
    size_t nA = (((size_t)N + 1) + 63) & ~(size_t)63;
    int* off   = (int*)d_ws;       // N+1
    int* deg   = off + nA;         // N
    int* cur   = deg + nA;         // N
    int* col_s = cur + nA;         // E + COL_PAD

    init_counts_kernel<<<(N + 255) / 256, 256, 0, stream>>>(deg, cur, col_s, N, E);
    count_kernel<<<(E + 255) / 256, 256, 0, stream>>>(row, deg, E);
    scan_kernel<<<1, 1024, 0, stream>>>(deg, off, N);
    fill_kernel<<<(E + 255) / 256, 256, 0, stream>>>(row, col, off, cur, col_s, E);

    dim3 block(32, 4);                    // 4 wave32s: one per 32-feature slab
    dim3 grid((N + 15) / 16);             // node tiles
    gather_wmma_kernel<<<grid, block, 0, stream>>>(x, off, col_s, rate, out, N);
}